// GraphToEmbedding_38963943309608
// MI455X (gfx1250) — compile-verified
//
#include <hip/hip_runtime.h>
#include <math.h>

// ---------------------------------------------------------------------------
// GraphToEmbedding on MI455X (gfx1250): GCN+GAT+2xGCN+Linear
// Dense GEMMs use V_WMMA_F32_16X16X4_F32 (fp32 matrix core, wave32).
// ---------------------------------------------------------------------------

typedef __attribute__((ext_vector_type(2))) float v2f;
typedef __attribute__((ext_vector_type(8))) float v8f;

#define HDIM 128
#define NEG_SLOPE 0.2f

__device__ __forceinline__ float leaky_relu(float x) {
  return x > 0.f ? x : NEG_SLOPE * x;
}

__device__ __forceinline__ void atomicMaxF(float* addr, float val) {
  unsigned int* ua = (unsigned int*)addr;
  unsigned int old = *ua;
  while (__uint_as_float(old) < val) {
    unsigned int assumed = old;
    old = atomicCAS(ua, assumed, __float_as_uint(val));
    if (old == assumed) break;
  }
}

// ---------------------------------------------------------------------------
// WMMA GEMM: Y[nrows x HOUT] = X[nrows x K] @ W[K x HOUT] (+ bias)
// One wave computes one 16-row tile across all HOUT/16 col tiles.
// W is staged in LDS. fp32 accumulation via v_wmma_f32_16x16x4_f32.
// ---------------------------------------------------------------------------
template <int K, int HOUT>
__global__ __launch_bounds__(256) void gemm_wmma_kernel(
    const float* __restrict__ X, const float* __restrict__ W,
    const float* __restrict__ bias, float* __restrict__ Y, int nrows) {
  __shared__ float sW[K * HOUT];
  for (int i = threadIdx.x; i < K * HOUT; i += 256) sW[i] = W[i];
  __syncthreads();

  const int lane = threadIdx.x & 31;
  const int wv = threadIdx.x >> 5;
  const int rt = blockIdx.x * 8 + wv;  // 16-row tile index
  if (rt * 16 >= nrows) return;

  const int m = lane & 15;       // row within tile (A) / col within tile (B)
  const int hk = lane >> 4;      // K-pair select: 0 -> {k,k+1}, 1 -> {k+2,k+3}
  const int koff = hk * 2;

  constexpr int CT = HOUT / 16;
  v8f acc[CT];
#pragma unroll
  for (int ct = 0; ct < CT; ++ct)
#pragma unroll
    for (int r = 0; r < 8; ++r) acc[ct][r] = 0.f;

  const float* xr = X + (size_t)(rt * 16 + m) * K;
  for (int k = 0; k < K; k += 4) {
    v2f a;
    a.x = xr[k + koff];
    a.y = xr[k + koff + 1];
    const float* w0 = &sW[(k + koff) * HOUT + m];
    const float* w1 = &sW[(k + koff + 1) * HOUT + m];
#pragma unroll
    for (int ct = 0; ct < CT; ++ct) {
      v2f b;
      b.x = w0[ct * 16];
      b.y = w1[ct * 16];
      acc[ct] = __builtin_amdgcn_wmma_f32_16x16x4_f32(
          false, a, false, b, (short)0, acc[ct], false, false);
    }
  }

  // C/D layout: VGPR r -> M = r + 8*hk, N = lane%16
  const int rbase = rt * 16 + hk * 8;
#pragma unroll
  for (int ct = 0; ct < CT; ++ct) {
    const int n = ct * 16 + m;
    const float bv = bias ? bias[n] : 0.f;
#pragma unroll
    for (int r = 0; r < 8; ++r)
      Y[(size_t)(rbase + r) * HOUT + n] = acc[ct][r] + bv;
  }
}

// ---------------------------------------------------------------------------
// Elementwise / graph kernels
// ---------------------------------------------------------------------------
__global__ void fill_kernel(float* p, float v, int n) {
  int i = blockIdx.x * 256 + threadIdx.x;
  if (i < n) p[i] = v;
}

// deg[col[e]] += (w ? w[e] : 1)
__global__ void edge_deg_kernel(const int* __restrict__ col,
                                const float* __restrict__ w,
                                float* __restrict__ deg, int E) {
  int e = blockIdx.x * 256 + threadIdx.x;
  if (e < E) atomicAdd(&deg[col[e]], w ? w[e] : 1.0f);
}

__global__ void rsqrt_kernel(float* d, int n) {
  int i = blockIdx.x * 256 + threadIdx.x;
  if (i < n) d[i] = 1.0f / sqrtf(d[i]);  // deg >= 1 always (self loop)
}

// acc[i,:] = xw[i,:] * dinv[i]^2   (self-loop term, weight 1.0 in all layers)
__global__ void self_init_kernel(float* __restrict__ acc,
                                 const float* __restrict__ xw,
                                 const float* __restrict__ dinv, int nchunks) {
  int t = blockIdx.x * 256 + threadIdx.x;
  if (t >= nchunks) return;
  int node = t >> 5;  // 32 float4 chunks per node (H=128)
  float di = dinv[node];
  float s = di * di;
  float4 v = ((const float4*)xw)[t];
  float4 o = {v.x * s, v.y * s, v.z * s, v.w * s};
  ((float4*)acc)[t] = o;
}

// acc[col,:] += xw[row,:] * dinv[row]*w_e*dinv[col]   (w_e = alpha or 1)
__global__ void scatter_kernel(const int* __restrict__ row,
                               const int* __restrict__ col,
                               const float* __restrict__ xw,
                               const float* __restrict__ dinv,
                               const float* __restrict__ alpha,
                               float* __restrict__ acc, int E) {
  int t = blockIdx.x * 256 + threadIdx.x;
  int e = t >> 5;
  if (e >= E) return;
  int c = t & 31;
  int r = row[e], d = col[e];
  float nrm = dinv[r] * dinv[d] * (alpha ? alpha[e] : 1.0f);
  float4 v = *(const float4*)(xw + (size_t)r * HDIM + c * 4);
  float* dst = acc + (size_t)d * HDIM + c * 4;
  atomicAdd(dst + 0, v.x * nrm);
  atomicAdd(dst + 1, v.y * nrm);
  atomicAdd(dst + 2, v.z * nrm);
  atomicAdd(dst + 3, v.w * nrm);
}

__global__ void relu_bias_kernel(float* __restrict__ xout,
                                 const float* __restrict__ acc,
                                 const float* __restrict__ b, int total) {
  int t = blockIdx.x * 256 + threadIdx.x;
  if (t >= total) return;
  int h = t & (HDIM - 1);
  xout[t] = fmaxf(acc[t] + b[h], 0.f);
}

// vgs = Wg @ att_src, vgd = Wg @ att_dst  (fold GAT transform into vectors)
__global__ void att_vec_kernel(const float* __restrict__ Wg,
                               const float* __restrict__ as_,
                               const float* __restrict__ ad_,
                               float* __restrict__ vgs,
                               float* __restrict__ vgd) {
  int h = threadIdx.x;  // 128 threads
  float s = 0.f, d = 0.f;
  for (int j = 0; j < HDIM; ++j) {
    float w = Wg[j * HDIM + h];
    s += w * as_[j];
    d += w * ad_[j];
  }
  vgs[h] = s;
  vgd[h] = d;
}

// a_s[i] = dot(x0[i], vgs), a_d[i] = dot(x0[i], vgd). One wave per node.
__global__ void a_sd_kernel(const float* __restrict__ x,
                            const float* __restrict__ vgs,
                            const float* __restrict__ vgd,
                            float* __restrict__ a_s, float* __restrict__ a_d,
                            int N) {
  int node = blockIdx.x * 8 + (threadIdx.x >> 5);
  int lane = threadIdx.x & 31;
  if (node >= N) return;
  float4 xv = *(const float4*)(x + (size_t)node * HDIM + lane * 4);
  float4 sv = *(const float4*)(vgs + lane * 4);
  float4 dv = *(const float4*)(vgd + lane * 4);
  float ps = xv.x * sv.x + xv.y * sv.y + xv.z * sv.z + xv.w * sv.w;
  float pd = xv.x * dv.x + xv.y * dv.y + xv.z * dv.z + xv.w * dv.w;
#pragma unroll
  for (int o = 16; o > 0; o >>= 1) {
    ps += __shfl_xor(ps, o, 32);
    pd += __shfl_xor(pd, o, 32);
  }
  if (lane == 0) {
    a_s[node] = ps;
    a_d[node] = pd;
  }
}

// m[i] = leaky(a_s[i] + a_d[i])  (self-loop edge seeds segment max)
__global__ void m_init_kernel(const float* a_s, const float* a_d, float* m,
                              int N) {
  int i = blockIdx.x * 256 + threadIdx.x;
  if (i < N) m[i] = leaky_relu(a_s[i] + a_d[i]);
}

__global__ void edge_e_max_kernel(const int* __restrict__ row,
                                  const int* __restrict__ col,
                                  const float* __restrict__ a_s,
                                  const float* __restrict__ a_d,
                                  float* __restrict__ ee, float* __restrict__ m,
                                  int E) {
  int e = blockIdx.x * 256 + threadIdx.x;
  if (e >= E) return;
  float v = leaky_relu(a_s[row[e]] + a_d[col[e]]);
  ee[e] = v;
  atomicMaxF(&m[col[e]], v);
}

// s[i] = exp(e_self - m)  (self-loop term of softmax denominator)
__global__ void s_init_kernel(const float* a_s, const float* a_d,
                              const float* m, float* s, int N) {
  int i = blockIdx.x * 256 + threadIdx.x;
  if (i < N) s[i] = expf(leaky_relu(a_s[i] + a_d[i]) - m[i]);
}

__global__ void edge_exp_sum_kernel(const int* __restrict__ col,
                                    const float* __restrict__ ee,
                                    const float* __restrict__ m,
                                    float* __restrict__ s, int E) {
  int e = blockIdx.x * 256 + threadIdx.x;
  if (e >= E) return;
  atomicAdd(&s[col[e]], expf(ee[e] - m[col[e]]));
}

__global__ void alpha_kernel(const int* __restrict__ col,
                             const float* __restrict__ ee,
                             const float* __restrict__ m,
                             const float* __restrict__ s,
                             float* __restrict__ alpha, int E) {
  int e = blockIdx.x * 256 + threadIdx.x;
  if (e >= E) return;
  int c = col[e];
  alpha[e] = expf(ee[e] - m[c]) / s[c];
}

// ---------------------------------------------------------------------------
extern "C" void kernel_launch(void* const* d_in, const int* in_sizes, int n_in,
                              void* d_out, int out_size, void* d_ws,
                              size_t ws_size, hipStream_t stream) {
  const float* x = (const float*)d_in[0];
  const int* ei = (const int*)d_in[1];
  // d_in[2] (batch) unused: equal-size sorted graphs -> output is a reshape
  const float* W0 = (const float*)d_in[3];
  const float* b0 = (const float*)d_in[4];
  const float* Wg = (const float*)d_in[5];
  const float* atts = (const float*)d_in[6];
  const float* attd = (const float*)d_in[7];
  const float* W1 = (const float*)d_in[8];
  const float* b1 = (const float*)d_in[9];
  const float* W2 = (const float*)d_in[10];
  const float* b2 = (const float*)d_in[11];
  const float* Wr = (const float*)d_in[12];
  const float* br = (const float*)d_in[13];
  float* out = (float*)d_out;

  const int N = in_sizes[2];
  const int E = in_sizes[1] / 2;
  const int* row = ei;      // edge_index[0]
  const int* col = ei + E;  // edge_index[1]

  float* ws = (float*)d_ws;
  const size_t NH = (size_t)N * HDIM;
  float* xw = ws;             // [N,128] GEMM output
  float* acc = xw + NH;       // [N,128] aggregation accumulator
  float* xbuf = acc + NH;     // [N,128] x0 -> x1 -> x2 (ping in place)
  float* dinv0 = xbuf + NH;   // [N]
  float* dinv1 = dinv0 + N;   // [N]
  float* a_s = dinv1 + N;     // [N]
  float* a_d = a_s + N;       // [N]
  float* mmax = a_d + N;      // [N]
  float* ssum = mmax + N;     // [N]
  float* alpha = ssum + N;    // [E]
  float* eedge = alpha + E;   // [E]
  float* vgs = eedge + E;     // [128]
  float* vgd = vgs + HDIM;    // [128]

  const dim3 blk(256);
  const int gN = (N + 255) / 256;
  const int gE = (E + 255) / 256;
  const int gNH4 = (int)((NH / 4 + 255) / 256);
  const int gE32 = (int)(((size_t)E * 32 + 255) / 256);
  const int gNH = (int)((NH + 255) / 256);
  const int gGemm = (N + 127) / 128;  // 8 waves/block, 16 rows/wave

  // ---- GCN layer 0 (edge weight 1, self loop weight 1) ----
  fill_kernel<<<gN, blk, 0, stream>>>(dinv0, 1.0f, N);
  edge_deg_kernel<<<gE, blk, 0, stream>>>(col, nullptr, dinv0, E);
  rsqrt_kernel<<<gN, blk, 0, stream>>>(dinv0, N);
  gemm_wmma_kernel<64, 128><<<gGemm, blk, 0, stream>>>(x, W0, nullptr, xw, N);
  self_init_kernel<<<gNH4, blk, 0, stream>>>(acc, xw, dinv0, (int)(NH / 4));
  scatter_kernel<<<gE32, blk, 0, stream>>>(row, col, xw, dinv0, nullptr, acc, E);
  relu_bias_kernel<<<gNH, blk, 0, stream>>>(xbuf, acc, b0, (int)NH);

  // ---- GAT attention (xg@att folded: a = x0 @ (Wg@att)) ----
  att_vec_kernel<<<1, HDIM, 0, stream>>>(Wg, atts, attd, vgs, vgd);
  a_sd_kernel<<<(N + 7) / 8, blk, 0, stream>>>(xbuf, vgs, vgd, a_s, a_d, N);
  m_init_kernel<<<gN, blk, 0, stream>>>(a_s, a_d, mmax, N);
  edge_e_max_kernel<<<gE, blk, 0, stream>>>(row, col, a_s, a_d, eedge, mmax, E);
  s_init_kernel<<<gN, blk, 0, stream>>>(a_s, a_d, mmax, ssum, N);
  edge_exp_sum_kernel<<<gE, blk, 0, stream>>>(col, eedge, mmax, ssum, E);
  alpha_kernel<<<gE, blk, 0, stream>>>(col, eedge, mmax, ssum, alpha, E);

  // ---- GCN layer 1 (edge weight alpha, self weight forced 1.0) ----
  fill_kernel<<<gN, blk, 0, stream>>>(dinv1, 1.0f, N);
  edge_deg_kernel<<<gE, blk, 0, stream>>>(col, alpha, dinv1, E);
  rsqrt_kernel<<<gN, blk, 0, stream>>>(dinv1, N);
  gemm_wmma_kernel<128, 128><<<gGemm, blk, 0, stream>>>(xbuf, W1, nullptr, xw, N);
  self_init_kernel<<<gNH4, blk, 0, stream>>>(acc, xw, dinv1, (int)(NH / 4));
  scatter_kernel<<<gE32, blk, 0, stream>>>(row, col, xw, dinv1, alpha, acc, E);
  relu_bias_kernel<<<gNH, blk, 0, stream>>>(xbuf, acc, b1, (int)NH);

  // ---- GCN layer 2 (same weights/norm as layer 1) ----
  gemm_wmma_kernel<128, 128><<<gGemm, blk, 0, stream>>>(xbuf, W2, nullptr, xw, N);
  self_init_kernel<<<gNH4, blk, 0, stream>>>(acc, xw, dinv1, (int)(NH / 4));
  scatter_kernel<<<gE32, blk, 0, stream>>>(row, col, xw, dinv1, alpha, acc, E);
  relu_bias_kernel<<<gNH, blk, 0, stream>>>(xbuf, acc, b2, (int)NH);

  // ---- readout: out = x2 @ Wr + br, written straight to d_out (reshape) ----
  gemm_wmma_kernel<128, 64><<<gGemm, blk, 0, stream>>>(xbuf, Wr, br, out, N);
}